// NodeSelfAttention_4698694222004
// MI455X (gfx1250) — compile-verified
//
#include <hip/hip_runtime.h>

// MI455X / gfx1250, wave32. Matrix math via V_WMMA_F32_16X16X4_F32; the
// 268 MB edge stream is double-buffered into LDS via GLOBAL_LOAD_ASYNC_TO_LDS.
typedef float v2f __attribute__((ext_vector_type(2)));
typedef float v8f __attribute__((ext_vector_type(8)));
typedef int v4i_vs __attribute__((vector_size(16)));  // matches builtin prototype
typedef __attribute__((address_space(1))) v4i_vs* as1_v4i_p;
typedef __attribute__((address_space(3))) v4i_vs* as3_v4i_p;

#define WMMA_F32(a, b, c) \
  __builtin_amdgcn_wmma_f32_16x16x4_f32(false, (a), false, (b), (short)0, (c), false, false)

#if __has_builtin(__builtin_amdgcn_global_load_async_to_lds_b128) && \
    __has_builtin(__builtin_amdgcn_s_wait_asynccnt)
#define ASYNC_LDS 1
#else
#define ASYNC_LDS 0
#endif

#define NN 2048
#define DD 256
#define EE 16
#define HH 8
#define QK 32
#define VV 32
#define OUTD 256

// LDS edge tile geometry (floats): 16-byte aligned slots, bank-conflict padded.
#define EM_STRIDE 20     // per-m slot: 16 data + 4 pad (mult of 4 => b128 aligned)
#define EN_STRIDE 1284   // per-n row: 64*20 + 4 pad  (1284 % 64 == 4, odd multiple/4)

// ---------------------------------------------------------------------------
// K1: q = ne*Wq, kT = (ne*Wk)^T, v = ne*Wv   via f32 WMMA.
// grid = (32, 24): y = h*3+which; block = 128 (4 waves, 16 rows each).
// ---------------------------------------------------------------------------
__global__ __launch_bounds__(128) void proj_kernel(
    const float* __restrict__ ne,   // [2048][256]
    const float* __restrict__ qt,   // [8][256][32]
    const float* __restrict__ kt,   // [8][256][32]
    const float* __restrict__ vt,   // [8][256][32]
    float* __restrict__ qbuf,       // [8][2048][32]
    float* __restrict__ kTbuf,      // [8][32][2048]
    float* __restrict__ vbuf)       // [8][2048][32]
{
  const int h = blockIdx.y / 3, which = blockIdx.y % 3;
  const int wave = threadIdx.x >> 5;
  const int lane = threadIdx.x & 31;
  const int half = lane >> 4, l16 = lane & 15;
  const int koff = half ? 2 : 0;
  const int n0 = blockIdx.x * 64 + wave * 16;

  const float* W = (which == 0 ? qt : (which == 1 ? kt : vt)) + (size_t)h * DD * QK;

  v8f acc0 = {}, acc1 = {};
  for (int k0 = 0; k0 < DD; k0 += 4) {
    v2f a;
    const float* ap = ne + (size_t)(n0 + l16) * DD + k0 + koff;
    a.x = ap[0]; a.y = ap[1];
    v2f b0, b1;
    const float* bp = W + (size_t)(k0 + koff) * QK;
    b0.x = bp[l16];      b0.y = bp[QK + l16];
    b1.x = bp[16 + l16]; b1.y = bp[QK + 16 + l16];
    acc0 = WMMA_F32(a, b0, acc0);
    acc1 = WMMA_F32(a, b1, acc1);
  }
#pragma unroll
  for (int r = 0; r < 8; ++r) {
    const int M = r + half * 8;
    if (which == 1) {  // transposed store: kT[h][q][n]
      kTbuf[(size_t)(h * QK + l16) * NN + n0 + M]      = acc0[r];
      kTbuf[(size_t)(h * QK + 16 + l16) * NN + n0 + M] = acc1[r];
    } else {
      float* o = (which == 0 ? qbuf : vbuf);
      o[(size_t)(h * NN + n0 + M) * QK + l16]      = acc0[r];
      o[(size_t)(h * NN + n0 + M) * QK + 16 + l16] = acc1[r];
    }
  }
}

// ---------------------------------------------------------------------------
// K2: flash-style fused attention. 128 blocks (16 query rows each),
// 8 waves = 8 heads. One async, double-buffered pass over edge_embeddings.
// ---------------------------------------------------------------------------
#define L_E0   0                          // 16*1284 = 20544
#define L_E1   20544                      // second edge buffer
#define L_EQ   41088                      // [8][16][16]        -> +2048
#define L_P    43136                      // [8][16*65]         -> +8320
#define L_SCL  51456                      // [8][16]            -> +128
#define L_ACCE 51584                      // [8][16][16]        -> +2048
#define L_AV   53632                      // [16][257]          -> +4112
#define L_TOT  57744                      // floats -> 230976 bytes (<320KB/WGP)

__device__ __forceinline__ void issue_edge_tile(const float* __restrict__ edge,
                                                float* sE, int n0, int m0, int tid) {
#pragma unroll
  for (int i = 0; i < 16; ++i) {
    const int nn = i, mm = tid >> 2, e4 = (tid & 3) * 4;
    const float* g = edge + ((size_t)(n0 + nn) * NN + m0 + mm) * EE + e4;
    float* l = sE + nn * EN_STRIDE + mm * EM_STRIDE + e4;
#if ASYNC_LDS
    __builtin_amdgcn_global_load_async_to_lds_b128(
        (as1_v4i_p)(void*)g, (as3_v4i_p)(void*)l, 0, 0);
#else
    float4 d = *(const float4*)g;
    l[0] = d.x; l[1] = d.y; l[2] = d.z; l[3] = d.w;
#endif
  }
}

__device__ __forceinline__ void edge_async_wait() {
#if ASYNC_LDS
  __builtin_amdgcn_s_wait_asynccnt(0);
#endif
}

__global__ __launch_bounds__(256) void attn_kernel(
    const float* __restrict__ edge,    // [2048][2048][16]
    const float* __restrict__ ektens,  // [16][32]
    const float* __restrict__ evtens,  // [16][32]
    const float* __restrict__ otens,   // [8][32][256] flattened [256][256]
    const float* __restrict__ qbuf,    // [8][2048][32]
    const float* __restrict__ kTbuf,   // [8][32][2048]
    const float* __restrict__ vbuf,    // [8][2048][32]
    float* __restrict__ out)           // [2048][256]
{
  extern __shared__ float smem[];
  float* sE0   = smem + L_E0;
  float* sE1   = smem + L_E1;
  float* sEq   = smem + L_EQ;
  float* sP    = smem + L_P;
  float* sScl  = smem + L_SCL;
  float* sAccE = smem + L_ACCE;
  float* sAV   = smem + L_AV;

  const int tid  = threadIdx.x;
  const int h    = tid >> 5;
  const int lane = tid & 31;
  const int half = lane >> 4, l16 = lane & 15;
  const int koff = half ? 2 : 0;
  const int n0   = blockIdx.x * 16;
  const float scale = 0.17677669529663687f;  // 1/sqrt(32)

  // kick off tile 0 before doing any prologue math (latency hiding)
  issue_edge_tile(edge, sE0, n0, 0, tid);

  // --- prologue: Q A-fragments (kept in regs) and eq[h][n][e] = q . ekt^T ---
  v2f aq[8];
#pragma unroll
  for (int t = 0; t < 8; ++t) {
    const float* p = qbuf + (size_t)(h * NN + n0 + l16) * QK + 4 * t + koff;
    aq[t].x = p[0]; aq[t].y = p[1];
  }
  {
    float qrow[QK];
    const float* p = qbuf + (size_t)(h * NN + n0 + l16) * QK;
#pragma unroll
    for (int q = 0; q < QK; ++q) qrow[q] = p[q];
#pragma unroll
    for (int j = 0; j < 8; ++j) {
      const int e = half * 8 + j;
      float s = 0.f;
#pragma unroll
      for (int q = 0; q < QK; ++q) s += qrow[q] * ektens[e * QK + q];
      sEq[(h * 16 + l16) * 16 + e] = s;   // same-wave producer/consumer
    }
  }

  float rmax[8], rsum[8], acce[8];
#pragma unroll
  for (int r = 0; r < 8; ++r) { rmax[r] = -3.0e38f; rsum[r] = 0.f; acce[r] = 0.f; }
  v8f accv0 = {}, accv1 = {};

  for (int it = 0; it < NN / 64; ++it) {
    const int m0 = it * 64;
    float* sEc = (it & 1) ? sE1 : sE0;   // tile being consumed
    float* sEn = (it & 1) ? sE0 : sE1;   // tile being prefetched

    edge_async_wait();   // this wave's share of tile `it` is in LDS
    __syncthreads();     // all waves done loading tile `it` AND reading tile `it-1`
    if (it + 1 < NN / 64) issue_edge_tile(edge, sEn, n0, m0 + 64, tid);

    // ---- logits S[16n x 64m] for head h: WMMA(Q K^T) + VALU edge term ----
    v8f sfr[4];
#pragma unroll
    for (int s = 0; s < 4; ++s) {
      v8f acc = {};
#pragma unroll
      for (int t = 0; t < 8; ++t) {  // K = 32
        v2f b;
        const float* p = kTbuf + (size_t)(h * QK + 4 * t + koff) * NN + m0 + s * 16 + l16;
        b.x = p[0]; b.y = p[NN];
        acc = WMMA_F32(aq[t], b, acc);
      }
#pragma unroll
      for (int r = 0; r < 8; ++r) {
        const int nn = r + half * 8;
        const float* ep = sEc + nn * EN_STRIDE + (s * 16 + l16) * EM_STRIDE;
        const float* qp = sEq + (h * 16 + nn) * 16;
        float se = 0.f;
#pragma unroll
        for (int e = 0; e < EE; ++e) se += ep[e] * qp[e];
        acc[r] = (acc[r] + se) * scale;
      }
      sfr[s] = acc;
    }

    // ---- online softmax (rows live per 16-lane half in D layout) ----
#pragma unroll
    for (int r = 0; r < 8; ++r) {
      float mx = fmaxf(fmaxf(sfr[0][r], sfr[1][r]), fmaxf(sfr[2][r], sfr[3][r]));
#pragma unroll
      for (int d = 1; d < 16; d <<= 1) mx = fmaxf(mx, __shfl_xor(mx, d, 32));
      const float nm = fmaxf(rmax[r], mx);
      const float corr = __expf(rmax[r] - nm);
      rmax[r] = nm;
      float rs = 0.f;
#pragma unroll
      for (int s = 0; s < 4; ++s) {
        const float pv = __expf(sfr[s][r] - nm);
        sfr[s][r] = pv; rs += pv;
      }
#pragma unroll
      for (int d = 1; d < 16; d <<= 1) rs += __shfl_xor(rs, d, 32);
      rsum[r] = rsum[r] * corr + rs;
      accv0[r] *= corr; accv1[r] *= corr;
      if (l16 == 0) sScl[h * 16 + r + half * 8] = corr;  // per-row corr broadcast
    }

    // stash P (wave-private) for layout transpose + edge accumulation
#pragma unroll
    for (int s = 0; s < 4; ++s)
#pragma unroll
      for (int r = 0; r < 8; ++r)
        sP[h * 1040 + (r + half * 8) * 65 + s * 16 + l16] = sfr[s][r];

    {  // rescale running edge stats (lane keys: n=l16, e=half*8+j)
      const float corr = sScl[h * 16 + l16];
#pragma unroll
      for (int j = 0; j < 8; ++j) acce[j] *= corr;
    }

    // ---- P · V via WMMA (K = 64) ----
#pragma unroll
    for (int t = 0; t < 16; ++t) {
      v2f a;
      const float* pp = sP + h * 1040 + l16 * 65 + 4 * t + koff;
      a.x = pp[0]; a.y = pp[1];
      const float* vp = vbuf + (size_t)(h * NN + m0 + 4 * t + koff) * VV + l16;
      v2f b0, b1;
      b0.x = vp[0];  b0.y = vp[VV];
      b1.x = vp[16]; b1.y = vp[VV + 16];
      accv0 = WMMA_F32(a, b0, accv0);
      accv1 = WMMA_F32(a, b1, accv1);
    }

    // ---- acc_e[n][e] += sum_m P[n][m]*edge[n][m][e]  (VALU, E=16) ----
    {
      const float* prow = sP + h * 1040 + l16 * 65;
      const float* erow = sEc + l16 * EN_STRIDE + half * 8;
#pragma unroll 4
      for (int mm = 0; mm < 64; ++mm) {
        const float pv = prow[mm];
        const float* ep = erow + mm * EM_STRIDE;
#pragma unroll
        for (int j = 0; j < 8; ++j) acce[j] += pv * ep[j];
      }
    }
  }  // m loop

  // --- epilogue: value = (accv + accE*evt)/l ; then fused output projection ---
#pragma unroll
  for (int j = 0; j < 8; ++j)
    sAccE[(h * 16 + l16) * 16 + half * 8 + j] = acce[j];

#pragma unroll
  for (int r = 0; r < 8; ++r) {
    const int nn = r + half * 8;
    const float invl = 1.0f / rsum[r];
    const float* ae = sAccE + (h * 16 + nn) * 16;
    float ev0 = 0.f, ev1 = 0.f;
#pragma unroll
    for (int e = 0; e < EE; ++e) {
      ev0 += ae[e] * evtens[e * VV + l16];
      ev1 += ae[e] * evtens[e * VV + 16 + l16];
    }
    sAV[nn * 257 + h * VV + l16]      = (accv0[r] + ev0) * invl;
    sAV[nn * 257 + h * VV + 16 + l16] = (accv1[r] + ev1) * invl;
  }
  __syncthreads();

  // out[16n][256] = AV[16n][256] * otens[256][256]; wave h does o-tiles 2h,2h+1
#pragma unroll
  for (int ot = 0; ot < 2; ++ot) {
    const int o0 = (h * 2 + ot) * 16;
    v8f acc = {};
#pragma unroll 4
    for (int t = 0; t < 64; ++t) {  // K = 256
      v2f a, b;
      const float* ap = sAV + l16 * 257 + 4 * t + koff;
      a.x = ap[0]; a.y = ap[1];
      const float* bp = otens + (size_t)(4 * t + koff) * OUTD + o0 + l16;
      b.x = bp[0]; b.y = bp[OUTD];
      acc = WMMA_F32(a, b, acc);
    }
#pragma unroll
    for (int r = 0; r < 8; ++r)
      out[(size_t)(n0 + r + half * 8) * OUTD + o0 + l16] = acc[r];
  }
}

// ---------------------------------------------------------------------------
extern "C" void kernel_launch(void* const* d_in, const int* in_sizes, int n_in,
                              void* d_out, int out_size, void* d_ws, size_t ws_size,
                              hipStream_t stream) {
  const float* edge = (const float*)d_in[0];
  const float* node = (const float*)d_in[1];
  const float* qt   = (const float*)d_in[2];
  const float* kt   = (const float*)d_in[3];
  const float* ekt  = (const float*)d_in[4];
  const float* nvt  = (const float*)d_in[5];
  const float* evt  = (const float*)d_in[6];
  const float* otn  = (const float*)d_in[7];

  float* ws    = (float*)d_ws;
  float* qbuf  = ws;                       // 8*2048*32 = 524288 floats (2 MB)
  float* kTbuf = ws + 524288;              // 2 MB
  float* vbuf  = ws + 2 * 524288;          // 2 MB

  proj_kernel<<<dim3(32, 24), 128, 0, stream>>>(node, qt, kt, nvt, qbuf, kTbuf, vbuf);

  const size_t smem = (size_t)L_TOT * sizeof(float);  // 230976 B, fits 320KB/WGP
  static int attr_set = 0;  // idempotent, capture-safe (not a stream op)
  if (!attr_set) {
    (void)hipFuncSetAttribute((const void*)attn_kernel,
                              hipFuncAttributeMaxDynamicSharedMemorySize, (int)smem);
    attr_set = 1;
  }
  attn_kernel<<<dim3(128), 256, smem, stream>>>(edge, ekt, evt, otn,
                                                qbuf, kTbuf, vbuf, (float*)d_out);
}